// SimpleCar_52243982188644
// MI455X (gfx1250) — compile-verified
//
#include <hip/hip_runtime.h>
#include <cstdint>

typedef float v4f __attribute__((ext_vector_type(4)));

#define TPB  256
#define EPT  4                 // elements per thread
#define TILE (TPB * EPT)       // 1024 elements per block

// Elementwise "rollout" (all reference cumsums are over singleton axes -> identity).
// Input  pa : [n][2]  (a0, a1) interleaved
// Output out: [n][6]  (x, y, yaw, vel, steer, accel)
__global__ __launch_bounds__(TPB) void SimpleCar_52243982188644_kernel(
    const float* __restrict__ state,
    const float* __restrict__ pa,
    float* __restrict__ out,
    long long n)
{
    __shared__ float tile[TILE * 2];   // 8 KB staging buffer

    const int t = threadIdx.x;
    const long long tile_base = (long long)blockIdx.x * TILE;
    const long long e0 = tile_base + (long long)t * EPT;

    // broadcast initial state (uniform -> scalar loads)
    const float s0 = state[0];
    const float s1 = state[1];
    const float s2 = state[2];
    const float s3 = state[3];
    const float s4 = state[4];
    const float s5 = state[5];

    v4f in0, in1;

    if (tile_base + TILE <= n) {
        // Fast path: stage this thread's 32 input bytes through LDS using the
        // CDNA5 async global->LDS data mover path (tracked by ASYNCcnt).
        const float* gsrc = pa + e0 * 2;                       // 32B per lane, coalesced
        unsigned lds_off = (unsigned)(uintptr_t)&tile[t * (EPT * 2)]; // low 32b = LDS offset
        asm volatile(
            "global_load_async_to_lds_b128 %0, %1, off\n\t"
            "global_load_async_to_lds_b128 %0, %1, off offset:16"
            :
            : "v"(lds_off), "v"(gsrc)
            : "memory");
        asm volatile("s_wait_asynccnt 0x0" ::: "memory");      // our async copies landed
        const v4f* lp = (const v4f*)&tile[t * (EPT * 2)];
        in0 = lp[0];
        in1 = lp[1];
    } else {
        // Generic tail path (not taken for the harness sizes: n % 1024 == 0)
        float buf[8];
#pragma unroll
        for (int k = 0; k < EPT; ++k) {
            long long e = e0 + k;
            float a = 0.0f, b = 0.0f;
            if (e < n) { a = pa[e * 2 + 0]; b = pa[e * 2 + 1]; }
            buf[2 * k + 0] = a;
            buf[2 * k + 1] = b;
        }
        in0 = *(const v4f*)&buf[0];
        in1 = *(const v4f*)&buf[4];
    }

    const float a0[EPT] = {in0.x, in0.z, in1.x, in1.z};
    const float a1[EPT] = {in0.y, in0.w, in1.y, in1.w};

    float res[EPT * 6];
#pragma unroll
    for (int k = 0; k < EPT; ++k) {
        const float steer = __builtin_fmaf(a0[k], 0.05f, s4);   // s4 + a0*DT
        const float accel = __builtin_fmaf(a1[k], 0.05f, s5);   // s5 + a1*DT
        const float vel   = __builtin_fmaf(accel, 0.5f,  s3);   // s3 + accel*10*DT
        const float delta = vel * 0.05f;                        // vel*DT
        const float yaw   = __builtin_fmaf(delta, steer, s2);   // K = steer*1.0
        float sn, cs;
        __sincosf(yaw, &sn, &cs);                               // v_sin_f32 / v_cos_f32
        res[k * 6 + 0] = __builtin_fmaf(delta, cs, s0);
        res[k * 6 + 1] = __builtin_fmaf(delta, sn, s1);
        res[k * 6 + 2] = yaw;
        res[k * 6 + 3] = vel;
        res[k * 6 + 4] = steer;
        res[k * 6 + 5] = accel;
    }

    if (e0 + EPT <= n) {
        // 4 elems * 24B = 96B per lane, 16B aligned -> 6x b128 non-temporal stores
        v4f* op = (v4f*)(out + e0 * 6);
#pragma unroll
        for (int k = 0; k < 6; ++k)
            __builtin_nontemporal_store(*(const v4f*)&res[4 * k], op + k);
    } else {
        for (int k = 0; k < EPT; ++k) {
            long long e = e0 + k;
            if (e < n)
                for (int c = 0; c < 6; ++c)
                    out[e * 6 + c] = res[k * 6 + c];
        }
    }
}

extern "C" void kernel_launch(void* const* d_in, const int* in_sizes, int n_in,
                              void* d_out, int out_size, void* d_ws, size_t ws_size,
                              hipStream_t stream) {
    (void)n_in; (void)out_size; (void)d_ws; (void)ws_size;
    const float* state = (const float*)d_in[0];   // 6 floats
    const float* pa    = (const float*)d_in[1];   // 8192*1024*2 floats
    float* out = (float*)d_out;                   // 8192*1024*6 floats

    const long long n = (long long)in_sizes[1] / 2;          // 8,388,608 elements
    const long long nblocks = (n + TILE - 1) / TILE;         // 8192 blocks
    SimpleCar_52243982188644_kernel<<<(unsigned)nblocks, TPB, 0, stream>>>(state, pa, out, n);
}